// TopographicalCorticalCell_36636071035134
// MI455X (gfx1250) — compile-verified
//
#include <hip/hip_runtime.h>
#include <hip/hip_bf16.h>

typedef __attribute__((ext_vector_type(2))) float v2f;
typedef __attribute__((ext_vector_type(8))) float v8f;

#if defined(__gfx1250__) && __has_builtin(__builtin_amdgcn_wmma_f32_16x16x4_f32)
#define USE_WMMA_TRANSPOSE 1
#else
#define USE_WMMA_TRANSPOSE 0
#endif

// ---------------------------------------------------------------------------
// Kernel 1: transpose x [B, N] -> xt [N, B]  (both directions coalesced)
// ---------------------------------------------------------------------------
__global__ __launch_bounds__(256) void transpose_k(const float* __restrict__ x,
                                                   float* __restrict__ xt,
                                                   int N, int B) {
    __shared__ float tile[32][33];
    const int n0 = blockIdx.x * 32;
    const int b0 = blockIdx.y * 32;
    const int tx = threadIdx.x;      // 0..31
    const int ty = threadIdx.y;      // 0..7
#pragma unroll
    for (int i = ty; i < 32; i += 8)
        tile[i][tx] = x[(size_t)(b0 + i) * N + n0 + tx];
    __syncthreads();
#pragma unroll
    for (int i = ty; i < 32; i += 8)
        xt[(size_t)(n0 + i) * B + b0 + tx] = tile[tx][i];
}

// ---------------------------------------------------------------------------
// Kernel 2: CSR row pointers from sorted COO rows (binary search per row)
// ---------------------------------------------------------------------------
__global__ __launch_bounds__(256) void rowptr_k(const int* __restrict__ rows,
                                                int* __restrict__ rp,
                                                int N, int E) {
    const int r = blockIdx.x * blockDim.x + threadIdx.x;
    if (r > N) return;
    int lo = 0, hi = E;
    while (lo < hi) {
        int mid = (lo + hi) >> 1;
        if (rows[mid] < r) lo = mid + 1; else hi = mid;
    }
    rp[r] = lo;   // first edge with row >= r ; rp[N] == E
}

// ---------------------------------------------------------------------------
// Kernel 3: SpMM.  One wave per row; lane l owns batch pair {2l, 2l+1}.
// 16 rows per block; epilogue transposes the 16x64 accumulator tile into
// out[B, N] layout using exact identity-WMMA (D = sum_c A_c x I_c).
// ---------------------------------------------------------------------------
#define TILE_ROWS 16
#define LDS_PITCH 72   // even (b64 LDS stores) ; 2*72 % 64 == 16 -> clean banks

__global__ __launch_bounds__(512) void spmm_k(const float* __restrict__ xt,
                                              const int* __restrict__ cols,
                                              const float* __restrict__ vals,
                                              const int* __restrict__ rp,
                                              const float* __restrict__ bias,
                                              float* __restrict__ out,
                                              int N) {
    __shared__ float T[TILE_ROWS][LDS_PITCH];   // [row_local][batch 0..63]

    const int wave = threadIdx.x >> 5;
    const int lane = threadIdx.x & 31;
    const int r0   = blockIdx.x * TILE_ROWS;
    const int r    = r0 + wave;

    const int e0 = rp[r];
    const int e1 = rp[r + 1];

    const float2* __restrict__ x2 = (const float2*)xt;  // row = 32 x float2
    float ax = 0.0f, ay = 0.0f;
    for (int e = e0; e < e1; ++e) {
        const int   c = cols[e];      // wave-uniform address -> single L0 req
        const float v = vals[e];
        const float2 xv = x2[(size_t)c * 32 + lane];    // 256B coalesced gather
        ax = fmaf(v, xv.x, ax);
        ay = fmaf(v, xv.y, ay);
    }
    const float bb = bias[r];
    T[wave][2 * lane]     = ax + bb;
    T[wave][2 * lane + 1] = ay + bb;
    __syncthreads();

#if USE_WMMA_TRANSPOSE
    // Waves 0..3 each transpose a 16(row) x 16(batch) subtile via WMMA.
    // A-layout (32-bit 16x4): lane%16 = M, lanes<16 hold K={0,1}, lanes>=16 K={2,3}.
    // B-layout (32-bit 4x16): lane%16 = N, lanes<16 hold K={0,1}, lanes>=16 K={2,3}.
    // With B_c = I[4c:4c+4, :]:  D[M][N] = T[N][j0+M]  (exact: one term per sum).
    if (wave < 4) {
        const int j0 = wave * 16;          // batch subtile base
        const int m  = lane & 15;
        const int hi = lane >> 4;          // 0: K={0,1}, 1: K={2,3}
        v8f d = {};
#pragma unroll
        for (int c = 0; c < 4; ++c) {
            const int k0 = 4 * c + (hi ? 2 : 0);
            v2f A, B;
            A.x = T[k0][j0 + m];           // A[M=m][K=k0]   = T[k0][j0+m]
            A.y = T[k0 + 1][j0 + m];       // A[M=m][K=k0+1] = T[k0+1][j0+m]
            B.x = (m == k0)     ? 1.0f : 0.0f;   // I[k0][N=m]
            B.y = (m == k0 + 1) ? 1.0f : 0.0f;   // I[k0+1][N=m]
            d = __builtin_amdgcn_wmma_f32_16x16x4_f32(
                    /*neg_a=*/false, A, /*neg_b=*/false, B,
                    /*c_mod=*/(short)0, d, /*reuse_a=*/false, /*reuse_b=*/false);
        }
        // D fragment: VGPR v, lanes 0-15 -> (M=v, N=lane); lanes 16-31 -> (M=v+8).
        // M = batch-within-subtile, N = row-within-tile -> coalesced 64B runs.
#pragma unroll
        for (int q = 0; q < 8; ++q)
            out[(size_t)(j0 + q + hi * 8) * N + r0 + m] = d[q];
    }
#else
    // Exact fallback: transposed write straight from LDS.
    for (int t = threadIdx.x; t < TILE_ROWS * 64; t += 512) {
        const int b  = t >> 4;
        const int rr = t & 15;
        out[(size_t)b * N + r0 + rr] = T[rr][b];
    }
#endif
}

// ---------------------------------------------------------------------------
// Host launcher
// ---------------------------------------------------------------------------
extern "C" void kernel_launch(void* const* d_in, const int* in_sizes, int n_in,
                              void* d_out, int out_size, void* d_ws, size_t ws_size,
                              hipStream_t stream) {
    const float* x       = (const float*)d_in[0];   // [B, N]
    const int*   indices = (const int*)  d_in[1];   // [2, E] : rows | cols
    const float* values  = (const float*)d_in[2];   // [E]
    const float* bias    = (const float*)d_in[3];   // [N]
    float*       out     = (float*)d_out;           // [B, N]

    const int N = in_sizes[3];            // 65536
    const int B = in_sizes[0] / N;        // 64
    const int E = in_sizes[2];            // ~2.16M

    const int* rows = indices;
    const int* cols = indices + E;

    float* xt = (float*)d_ws;                                   // N*B floats
    int*   rp = (int*)((char*)d_ws + sizeof(float) * (size_t)N * B);

    // 1) x -> xt
    dim3 tb(32, 8);
    dim3 tg((N + 31) / 32, (B + 31) / 32);
    transpose_k<<<tg, tb, 0, stream>>>(x, xt, N, B);

    // 2) CSR row pointers
    rowptr_k<<<(N + 1 + 255) / 256, 256, 0, stream>>>(rows, rp, N, E);

    // 3) SpMM + bias + transpose-out
    spmm_k<<<N / TILE_ROWS, 512, 0, stream>>>(xt, cols, values, rp, bias, out, N);
}